// GraphSAGE_51187420234379
// MI455X (gfx1250) — compile-verified
//
#include <hip/hip_runtime.h>
#include <hip/hip_bf16.h>

#define NN 50000
#define D  512

typedef __attribute__((ext_vector_type(2))) float v2f;
typedef __attribute__((ext_vector_type(8))) float v8f;

__device__ __forceinline__ int clampn(long long v) {
    int i = (int)v;
    return i < 0 ? 0 : (i >= NN ? NN - 1 : i);
}

// XOR-swizzled LDS index: element (r, c) of a 16x512 panel, c even-pair swizzle.
// Keeps 16 lanes reading column kk of rows 0..15 on distinct banks, 64KB total.
__device__ __forceinline__ int lds_idx(int r, int c) {
    int cpair = c >> 1;
    return r * D + (((cpair ^ (r << 1)) << 1) | (c & 1));
}

// ---- degree ----------------------------------------------------------------
__global__ void deg_kernel(const long long* __restrict__ dst,
                           float* __restrict__ deg, int nE) {
    int e = blockIdx.x * blockDim.x + threadIdx.x;
    if (e < nE) atomicAdd(&deg[clampn(dst[e])], 1.0f);
}

__global__ void invdeg_kernel(float* __restrict__ deg) {
    int n = blockIdx.x * blockDim.x + threadIdx.x;
    if (n < NN) deg[n] = 1.0f / fmaxf(deg[n], 1.0f);
}

// ---- edge scatter-add with fused mean scaling ------------------------------
// agg[dst] += feat[src] * invdeg[dst]  =>  agg holds the mean directly.
__global__ void __launch_bounds__(256)
scatter_kernel(const float* __restrict__ feat,
               const long long* __restrict__ src,
               const long long* __restrict__ dst,
               const float* __restrict__ invdeg,
               float* __restrict__ agg, int nE) {
    int q   = threadIdx.x & 127;   // 128 quads = 512 floats
    int sub = threadIdx.x >> 7;    // 0..1
    int e   = blockIdx.x * 2 + sub;
    if (e >= nE) return;
    int s = clampn(src[e]);
    int d = clampn(dst[e]);
    float rs = invdeg[d];
    float4 v = ((const float4*)(feat + (size_t)s * D))[q];
    float* ap = agg + (size_t)d * D + (size_t)q * 4;
    atomicAdd(ap + 0, v.x * rs);
    atomicAdd(ap + 1, v.y * rs);
    atomicAdd(ap + 2, v.z * rs);
    atomicAdd(ap + 3, v.w * rs);
}

// ---- SAGEConv GEMM: Out = relu( Agg@Wl + X@Wr + bl ) -----------------------
// One 1024-thread block (32 waves) per 16-row M-tile; wave w owns N-tile w.
// A panels staged once into LDS via async global->LDS copies (ASYNCcnt path),
// WMMA f32 16x16x4 feeds A from ds_load_b64, B from L2-resident weights.
__global__ void __launch_bounds__(1024)
sage_gemm(const float* __restrict__ Agg, const float* __restrict__ X,
          const float* __restrict__ Wl, const float* __restrict__ bl,
          const float* __restrict__ Wr,
          float* __restrict__ Out, int do_relu) {
    __shared__ float As[16 * D];   // 32KB
    __shared__ float Xs[16 * D];   // 32KB

    const int tid     = threadIdx.x;
    const int mTile   = blockIdx.x;        // 0..3124
    const int rowBase = mTile << 4;

    // Async stage: 2048 x 16B per panel, 1024 threads, 2 iters, no VGPR bounce.
#pragma unroll
    for (int i = 0; i < 2; ++i) {
        int idx = tid + (i << 10);         // 0..2047
        int r   = idx >> 7;                // 0..15
        int c   = (idx & 127) << 2;        // 0..508 step 4 (16B granules)
        unsigned la = (unsigned)(uintptr_t)(As + lds_idx(r, c)); // LDS byte offset
        unsigned lx = (unsigned)(uintptr_t)(Xs + lds_idx(r, c));
        unsigned long long ga =
            (unsigned long long)(uintptr_t)(Agg + (size_t)(rowBase + r) * D + c);
        unsigned long long gx =
            (unsigned long long)(uintptr_t)(X + (size_t)(rowBase + r) * D + c);
        asm volatile("global_load_async_to_lds_b128 %0, %1, off"
                     :: "v"(la), "v"(ga) : "memory");
        asm volatile("global_load_async_to_lds_b128 %0, %1, off"
                     :: "v"(lx), "v"(gx) : "memory");
    }
    asm volatile("s_wait_asynccnt 0x0" ::: "memory");
    __syncthreads();

    const int wave  = tid >> 5;            // = N-tile, 0..31
    const int lane  = tid & 31;
    const int l15   = lane & 15;
    const int khalf = lane >> 4;           // 0: K={0,1}, 1: K={2,3}
    const int col   = (wave << 4) + l15;

    v8f acc;
    const float bias = bl[col];
#pragma unroll
    for (int j = 0; j < 8; ++j) acc[j] = bias;

    __builtin_prefetch(Wl + col, 0, 0);    // global_prefetch_b8
    __builtin_prefetch(Wr + col, 0, 0);

#pragma unroll 4
    for (int k = 0; k < D; k += 4) {
        const int kk = k + (khalf << 1);
        v2f a = *(const v2f*)(As + lds_idx(l15, kk));   // ds_load_b64
        v2f b; b.x = Wl[(size_t)kk * D + col];
               b.y = Wl[(size_t)(kk + 1) * D + col];
        acc = __builtin_amdgcn_wmma_f32_16x16x4_f32(
                  false, a, false, b, (short)0, acc, false, false);
    }
#pragma unroll 4
    for (int k = 0; k < D; k += 4) {
        const int kk = k + (khalf << 1);
        v2f a = *(const v2f*)(Xs + lds_idx(l15, kk));
        v2f b; b.x = Wr[(size_t)kk * D + col];
               b.y = Wr[(size_t)(kk + 1) * D + col];
        acc = __builtin_amdgcn_wmma_f32_16x16x4_f32(
                  false, a, false, b, (short)0, acc, false, false);
    }

#pragma unroll
    for (int j = 0; j < 8; ++j) {
        float v = acc[j];
        if (do_relu) v = fmaxf(v, 0.0f);
        const int m = rowBase + (khalf << 3) + j;   // C layout: vgpr j, lane halves
        Out[(size_t)m * D + col] = v;
    }
}

// ---- final 512x2 head ------------------------------------------------------
__global__ void __launch_bounds__(256)
final_linear(const float* __restrict__ H, const float* __restrict__ Wlin,
             const float* __restrict__ blin, float* __restrict__ Out) {
    int idx = blockIdx.x * blockDim.x + threadIdx.x;   // over NN*2
    if (idx >= NN * 2) return;
    int n = idx >> 1, o = idx & 1;
    const float* h = H + (size_t)n * D;
    float s = blin[o];
#pragma unroll 4
    for (int k = 0; k < D; k += 4) {
        float4 hv = *(const float4*)(h + k);
        s += hv.x * Wlin[(k + 0) * 2 + o] + hv.y * Wlin[(k + 1) * 2 + o]
           + hv.z * Wlin[(k + 2) * 2 + o] + hv.w * Wlin[(k + 3) * 2 + o];
    }
    Out[idx] = s;
}

extern "C" void kernel_launch(void* const* d_in, const int* in_sizes, int n_in,
                              void* d_out, int out_size, void* d_ws, size_t ws_size,
                              hipStream_t stream) {
    const float*     x  = (const float*)d_in[0];
    const long long* ei = (const long long*)d_in[1];    // int64 per reference
    const float* Wl0 = (const float*)d_in[2];
    const float* bl0 = (const float*)d_in[3];
    const float* Wr0 = (const float*)d_in[4];
    const float* Wl1 = (const float*)d_in[5];
    const float* bl1 = (const float*)d_in[6];
    const float* Wr1 = (const float*)d_in[7];
    const float* Wl2 = (const float*)d_in[8];
    const float* bl2 = (const float*)d_in[9];
    const float* Wr2 = (const float*)d_in[10];
    const float* Wlin = (const float*)d_in[11];
    const float* blin = (const float*)d_in[12];

    const int nE = in_sizes[1] / 2;
    const long long* src = ei;         // edge_index[0]
    const long long* dst = ei + nE;    // edge_index[1]

    float* agg    = (float*)d_ws;                  // [NN, D]
    float* h0     = agg + (size_t)NN * D;          // [NN, D]
    float* h1     = h0  + (size_t)NN * D;          // [NN, D]
    float* invdeg = h1  + (size_t)NN * D;          // [NN]

    float* out2 = (float*)d_out;                   // [NN, 2]
    float* hfin = out2 + (size_t)NN * 2;           // [NN, D] (tuple output h)

    const size_t featBytes  = (size_t)NN * D * sizeof(float);
    const int scatterBlocks = (nE + 1) / 2;
    const dim3 ggrid(NN / 16, 1, 1);               // 3125 M-tiles, 32 waves each

    // degree -> 1/max(deg,1), computed once (before any scatter)
    hipMemsetAsync(invdeg, 0, NN * sizeof(float), stream);
    deg_kernel<<<(nE + 255) / 256, 256, 0, stream>>>(dst, invdeg, nE);
    invdeg_kernel<<<(NN + 255) / 256, 256, 0, stream>>>(invdeg);

    // layer 0
    hipMemsetAsync(agg, 0, featBytes, stream);
    scatter_kernel<<<scatterBlocks, 256, 0, stream>>>(x, src, dst, invdeg, agg, nE);
    sage_gemm<<<ggrid, 1024, 0, stream>>>(agg, x, Wl0, bl0, Wr0, h0, 1);

    // layer 1
    hipMemsetAsync(agg, 0, featBytes, stream);
    scatter_kernel<<<scatterBlocks, 256, 0, stream>>>(h0, src, dst, invdeg, agg, nE);
    sage_gemm<<<ggrid, 1024, 0, stream>>>(agg, h0, Wl1, bl1, Wr1, h1, 1);

    // layer 2 -> writes h directly into d_out tuple slot
    hipMemsetAsync(agg, 0, featBytes, stream);
    scatter_kernel<<<scatterBlocks, 256, 0, stream>>>(h1, src, dst, invdeg, agg, nE);
    sage_gemm<<<ggrid, 1024, 0, stream>>>(agg, h1, Wl2, bl2, Wr2, hfin, 1);

    // head
    final_linear<<<(NN * 2 + 255) / 256, 256, 0, stream>>>(hfin, Wlin, blin, out2);
}